// Cell_35150012350525
// MI455X (gfx1250) — compile-verified
//
#include <hip/hip_runtime.h>

typedef __attribute__((ext_vector_type(2))) float v2f;
typedef __attribute__((ext_vector_type(8))) float v8f;

#define NNODES 50000
#define KDIM 256
#define NCOLS 256
#define NSTRIPS (NNODES / 16)   // 3125, exact
#define KCHUNK 64               // K rows staged per LDS chunk
#define NPAIRS (KCHUNK / 2)     // 32 interleaved k-pair rows
#define PAIRSTRIDE 544          // 512 + 32 pad: 544 % 64 == 32 -> lane halves on disjoint banks
#define BN_EPS 1e-5f
#define ALPHA 0.1f
#define BETA_IM 0.40546510810816444f   // log(0.5/1 + 1)

// ---------------- prep: W_eff = beta*W_op + (1-beta)*I ; zero col stats ---
__global__ __launch_bounds__(256)
void prep_kernel(const float* __restrict__ Wop, float* __restrict__ Weff,
                 float* __restrict__ colsum, float* __restrict__ colsumsq) {
    int i = blockIdx.x * 256 + threadIdx.x;          // 0..65535
    int r = i >> 8, c = i & 255;
    float w = BETA_IM * Wop[i];
    if (r == c) w += (1.0f - BETA_IM);
    Weff[i] = w;
    if (i < 256) { colsum[i] = 0.0f; colsumsq[i] = 0.0f; }
}

// ---------------- fp32 WMMA GEMM: C[N,256] = A[N,256] @ B[256,256] ---------
// One wave -> 16 rows x 256 cols (16 accumulator tiles).
// B staged via LDS in k-pair-interleaved layout so each B operand is one
// aligned ds_load_b64 straight into an even VGPR pair (no mov shuffle).
__global__ __launch_bounds__(256)
void gemm_wmma_kernel(const float* __restrict__ A, const float* __restrict__ B,
                      float* __restrict__ C, int apply_relu) {
    __shared__ float Blds[NPAIRS * PAIRSTRIDE];      // 68 KB

    const int tid   = threadIdx.x;
    const int lane  = tid & 31;
    const int wave  = tid >> 5;
    const int strip = blockIdx.x * 8 + wave;
    const bool valid = (strip < NSTRIPS);
    const int row0  = strip * 16;
    const int m     = lane & 15;                     // A row / B,C col within tile
    const int half  = lane >> 4;                     // 0 or 1
    const int kbase = half * 2;                      // K sub-offset per lane half

    v8f acc[16];
#pragma unroll
    for (int ct = 0; ct < 16; ++ct)
#pragma unroll
        for (int r = 0; r < 8; ++r) acc[ct][r] = 0.0f;

    for (int chunk = 0; chunk < KDIM / KCHUNK; ++chunk) {
        // stage B rows [chunk*64, +64) interleaved by k-pairs:
        //   Blds[p*PAIRSTRIDE + n*2 + 0] = B[2p][n],  +1 = B[2p+1][n]
        const float4* Bg4 = (const float4*)(B + chunk * KCHUNK * NCOLS);
        for (int idx = tid; idx < NPAIRS * 16; idx += 256) {
            int p  = idx >> 4;                       // pair row 0..31
            int n4 = idx & 15;                       // group of 4 columns
            float4 r0 = Bg4[(2 * p)     * (NCOLS / 4) + n4];
            float4 r1 = Bg4[(2 * p + 1) * (NCOLS / 4) + n4];
            float* dst = &Blds[p * PAIRSTRIDE + n4 * 8];
            ((float4*)dst)[0] = make_float4(r0.x, r1.x, r0.y, r1.y);
            ((float4*)dst)[1] = make_float4(r0.z, r1.z, r0.w, r1.w);
        }
        __syncthreads();

        if (valid) {
            const float* Arow = A + (row0 + m) * KDIM + chunk * KCHUNK + kbase;
            for (int ks = 0; ks < KCHUNK; ks += 4) {
                v2f a = *(const v2f*)(Arow + ks);    // k, k+1 (k = chunk*64+ks+kbase)
                // pair row for this lane's K: (ks + kbase)/2 = ks/2 + half
                const v2f* bp = (const v2f*)&Blds[(ks / 2 + half) * PAIRSTRIDE];
#pragma unroll
                for (int ct = 0; ct < 16; ++ct) {
                    v2f b = bp[ct * 16 + m];         // single ds_load_b64
                    acc[ct] = __builtin_amdgcn_wmma_f32_16x16x4_f32(
                        false, a, false, b, (short)0, acc[ct], false, false);
                }
            }
        }
        __syncthreads();
    }

    if (valid) {
        const int rbase = 8 * half;                  // C/D: M = r + 8*(lane/16)
#pragma unroll
        for (int ct = 0; ct < 16; ++ct)
#pragma unroll
            for (int r = 0; r < 8; ++r) {
                float v = acc[ct][r];
                if (apply_relu) v = fmaxf(v, 0.0f);
                C[(row0 + rbase + r) * NCOLS + ct * 16 + m] = v;
            }
    }
}

// ---------------- per-column sum / sumsq (coalesced, grid-strided rows) ----
__global__ __launch_bounds__(256)
void col_stats_kernel(const float* __restrict__ H, float* __restrict__ colsum,
                      float* __restrict__ colsumsq) {
    const int c = threadIdx.x;
    float s = 0.0f, sq = 0.0f;
    for (int r = blockIdx.x; r < NNODES; r += gridDim.x) {
        float v = H[r * NCOLS + c];
        s += v; sq += v * v;
    }
    __hip_atomic_fetch_add(&colsum[c],   s,  __ATOMIC_RELAXED, __HIP_MEMORY_SCOPE_AGENT);
    __hip_atomic_fetch_add(&colsumsq[c], sq, __ATOMIC_RELAXED, __HIP_MEMORY_SCOPE_AGENT);
}

__global__ __launch_bounds__(256)
void finalize_stats_kernel(const float* __restrict__ colsum,
                           const float* __restrict__ colsumsq,
                           float* __restrict__ mu, float* __restrict__ rstd) {
    int c = threadIdx.x;
    float mean = colsum[c] * (1.0f / NNODES);
    float var  = colsumsq[c] * (1.0f / NNODES) - mean * mean;
    mu[c]   = mean;
    rstd[c] = rsqrtf(var + BN_EPS);
}

// -------- BN + ReLU in place; seed AGG = h (folds (1+eps)*h, eps=0) --------
__global__ __launch_bounds__(256)
void bn_relu_kernel(float* __restrict__ H, float* __restrict__ AGG,
                    const float* __restrict__ mu, const float* __restrict__ rstd,
                    const float* __restrict__ gamma, const float* __restrict__ beta) {
    int i = blockIdx.x * 256 + threadIdx.x;          // float4 index
    int base = i * 4;
    int c = base & (NCOLS - 1);
    float4 h = *(const float4*)(H + base);
    float4 o;
    o.x = fmaxf((h.x - mu[c + 0]) * rstd[c + 0] * gamma[c + 0] + beta[c + 0], 0.0f);
    o.y = fmaxf((h.y - mu[c + 1]) * rstd[c + 1] * gamma[c + 1] + beta[c + 1], 0.0f);
    o.z = fmaxf((h.z - mu[c + 2]) * rstd[c + 2] * gamma[c + 2] + beta[c + 2], 0.0f);
    o.w = fmaxf((h.w - mu[c + 3]) * rstd[c + 3] * gamma[c + 3] + beta[c + 3], 0.0f);
    *(float4*)(H + base)   = o;
    *(float4*)(AGG + base) = o;
}

// -------- edge aggregation: one wave per edge, 8 channels per lane ---------
__global__ __launch_bounds__(256)
void edge_agg_kernel(const float* __restrict__ H, const int* __restrict__ ei,
                     float* __restrict__ AGG, int E) {
    int t = blockIdx.x * 256 + threadIdx.x;
    int e = t >> 5, lane = t & 31;
    if (e >= E) return;
    int src = ei[e];
    int dst = ei[E + e];
    const float4* hs = (const float4*)(H + src * NCOLS);
    float4 v0 = hs[lane * 2];
    float4 v1 = hs[lane * 2 + 1];
    float* ap = AGG + dst * NCOLS + lane * 8;
    __hip_atomic_fetch_add(ap + 0, v0.x, __ATOMIC_RELAXED, __HIP_MEMORY_SCOPE_AGENT);
    __hip_atomic_fetch_add(ap + 1, v0.y, __ATOMIC_RELAXED, __HIP_MEMORY_SCOPE_AGENT);
    __hip_atomic_fetch_add(ap + 2, v0.z, __ATOMIC_RELAXED, __HIP_MEMORY_SCOPE_AGENT);
    __hip_atomic_fetch_add(ap + 3, v0.w, __ATOMIC_RELAXED, __HIP_MEMORY_SCOPE_AGENT);
    __hip_atomic_fetch_add(ap + 4, v1.x, __ATOMIC_RELAXED, __HIP_MEMORY_SCOPE_AGENT);
    __hip_atomic_fetch_add(ap + 5, v1.y, __ATOMIC_RELAXED, __HIP_MEMORY_SCOPE_AGENT);
    __hip_atomic_fetch_add(ap + 6, v1.z, __ATOMIC_RELAXED, __HIP_MEMORY_SCOPE_AGENT);
    __hip_atomic_fetch_add(ap + 7, v1.w, __ATOMIC_RELAXED, __HIP_MEMORY_SCOPE_AGENT);
}

// -------- support = (1-alpha)*(h+agg) + alpha*x0, in place on AGG ----------
__global__ __launch_bounds__(256)
void support_kernel(float* __restrict__ S, const float* __restrict__ X0) {
    int i = blockIdx.x * 256 + threadIdx.x;
    float4 s = ((const float4*)S)[i];
    float4 x = ((const float4*)X0)[i];
    float4 o;
    o.x = (1.0f - ALPHA) * s.x + ALPHA * x.x;
    o.y = (1.0f - ALPHA) * s.y + ALPHA * x.y;
    o.z = (1.0f - ALPHA) * s.z + ALPHA * x.z;
    o.w = (1.0f - ALPHA) * s.w + ALPHA * x.w;
    ((float4*)S)[i] = o;
}

extern "C" void kernel_launch(void* const* d_in, const int* in_sizes, int n_in,
                              void* d_out, int out_size, void* d_ws, size_t ws_size,
                              hipStream_t stream) {
    const float* s1    = (const float*)d_in[1];
    const float* x0    = (const float*)d_in[2];
    const float* Wpre  = (const float*)d_in[3];
    const float* gamma = (const float*)d_in[4];
    const float* betab = (const float*)d_in[5];
    const float* Wop   = (const float*)d_in[6];
    const int*   ei    = (const int*)d_in[7];
    const int E = in_sizes[7] / 2;
    float* out = (float*)d_out;

    float* ws       = (float*)d_ws;
    float* H        = ws;                               // N*256 floats
    float* AGG      = H + NNODES * NCOLS;               // N*256 floats
    float* Weff     = AGG + NNODES * NCOLS;             // 65536 floats
    float* colsum   = Weff + KDIM * NCOLS;              // 256
    float* colsumsq = colsum + 256;                     // 256
    float* mu       = colsumsq + 256;                   // 256
    float* rstd     = mu + 256;                         // 256

    const int gemm_grid = (NSTRIPS + 7) / 8;            // 391 blocks, 8 waves each
    const int elem4_grid = (NNODES * NCOLS) / (256 * 4);// 12500

    prep_kernel<<<256, 256, 0, stream>>>(Wop, Weff, colsum, colsumsq);
    gemm_wmma_kernel<<<gemm_grid, 256, 0, stream>>>(s1, Wpre, H, 0);
    col_stats_kernel<<<256, 256, 0, stream>>>(H, colsum, colsumsq);
    finalize_stats_kernel<<<1, 256, 0, stream>>>(colsum, colsumsq, mu, rstd);
    bn_relu_kernel<<<elem4_grid, 256, 0, stream>>>(H, AGG, mu, rstd, gamma, betab);
    edge_agg_kernel<<<(E * 32 + 255) / 256, 256, 0, stream>>>(H, ei, AGG, E);
    support_kernel<<<elem4_grid, 256, 0, stream>>>(AGG, x0);
    gemm_wmma_kernel<<<gemm_grid, 256, 0, stream>>>(AGG, Weff, out, 1);
}